// Decoder_1675037245972
// MI455X (gfx1250) — compile-verified
//
#include <hip/hip_runtime.h>
#include <hip/hip_bf16.h>
#include <math.h>

// Sizes (hardcoded from reference)
#define Bn 32
#define Tn 64
#define Sn 64
#define Hn 512
#define En 512
#define Vn 32000
#define H4 2048   // 4*H
#define H2 1024   // 2*H
#define H3 1536   // 3*H
#define MT 2048   // B*T

typedef __attribute__((ext_vector_type(2))) float v2f;
typedef __attribute__((ext_vector_type(8))) float v8f;

__device__ __forceinline__ v8f wmma4(v2f a, v2f b, v8f c) {
  // D = A(16x4,f32) x B(4x16,f32) + C(16x16,f32)
  return __builtin_amdgcn_wmma_f32_16x16x4_f32(false, a, false, b, (short)0, c,
                                               false, false);
}

__device__ __forceinline__ float sigm(float x) {
  return 1.0f / (1.0f + __expf(-x));
}

// ---------------------------------------------------------------------------
// Kernel 0: copy h0/c0 into step-0 slots of the state history
// ---------------------------------------------------------------------------
__global__ void k_init(const float* __restrict__ h0, const float* __restrict__ c0,
                       float* __restrict__ h_seq, float* __restrict__ c_seq) {
  int i = blockIdx.x * blockDim.x + threadIdx.x;
  if (i < Bn * Hn) { h_seq[i] = h0[i]; c_seq[i] = c0[i]; }
}

// ---------------------------------------------------------------------------
// Kernel 1: XW[m, :] = emb[inputs[m]] @ W_ih^T + b_ih + b_hh   (m = b*T + t)
// M=2048, N=2048, K=512. Wave computes a 32x64 tile (2x4 WMMA tiles):
// 6 b64 loads -> 8 WMMAs per k-step.
// grid (N/512=4, M/32=64), block 256 (8 waves).
// ---------------------------------------------------------------------------
__global__ void k_embed_xw(const int* __restrict__ inputs,
                           const float* __restrict__ emb,
                           const float* __restrict__ W_ih,
                           const float* __restrict__ b_ih,
                           const float* __restrict__ b_hh,
                           float* __restrict__ XW) {
  const int lane = threadIdx.x & 31, wid = threadIdx.x >> 5;
  const int ll = lane & 15, lh = lane >> 4;
  const int m0 = blockIdx.y * 32;
  const int n0 = blockIdx.x * 512 + wid * 64;

  const float* arow0 = emb + (size_t)inputs[m0 + ll] * En;
  const float* arow1 = emb + (size_t)inputs[m0 + 16 + ll] * En;
  const float* b0 = W_ih + (size_t)(n0 +  0 + ll) * En;
  const float* b1 = W_ih + (size_t)(n0 + 16 + ll) * En;
  const float* b2 = W_ih + (size_t)(n0 + 32 + ll) * En;
  const float* b3 = W_ih + (size_t)(n0 + 48 + ll) * En;

  v8f c00 = {}, c01 = {}, c02 = {}, c03 = {};
  v8f c10 = {}, c11 = {}, c12 = {}, c13 = {};
  for (int k = 0; k < En; k += 4) {
    const int kk = k + 2 * lh;
    v2f a0 = { arow0[kk], arow0[kk + 1] };
    v2f a1 = { arow1[kk], arow1[kk + 1] };
    v2f f0 = { b0[kk], b0[kk + 1] };
    v2f f1 = { b1[kk], b1[kk + 1] };
    v2f f2 = { b2[kk], b2[kk + 1] };
    v2f f3 = { b3[kk], b3[kk + 1] };
    c00 = wmma4(a0, f0, c00);  c01 = wmma4(a0, f1, c01);
    c02 = wmma4(a0, f2, c02);  c03 = wmma4(a0, f3, c03);
    c10 = wmma4(a1, f0, c10);  c11 = wmma4(a1, f1, c11);
    c12 = wmma4(a1, f2, c12);  c13 = wmma4(a1, f3, c13);
  }
#pragma unroll
  for (int i = 0; i < 4; ++i) {
    const v8f lo = (i == 0) ? c00 : (i == 1) ? c01 : (i == 2) ? c02 : c03;
    const v8f hi = (i == 0) ? c10 : (i == 1) ? c11 : (i == 2) ? c12 : c13;
    const int col = n0 + i * 16 + ll;
    const float bias = b_ih[col] + b_hh[col];
#pragma unroll
    for (int r = 0; r < 8; ++r) {
      XW[(size_t)(m0 + r + 8 * lh) * H4 + col] = lo[r] + bias;
      XW[(size_t)(m0 + 16 + r + 8 * lh) * H4 + col] = hi[r] + bias;
    }
  }
}

// ---------------------------------------------------------------------------
// Kernel 2 (per step t): gates = XW[b*T+t, :] + h_{t-1} @ W_hh^T ; LSTM cell.
// Each wave owns 16 batch rows x 16 h-columns and computes all 4 gate tiles
// (i,f,g,o at column offsets 0/512/1024/1536) so it can finish h,c locally.
// grid 8 blocks x 256 threads = 64 waves = (2 m-tiles) x (32 h-tiles).
// ---------------------------------------------------------------------------
__global__ void k_lstm_step(const float* __restrict__ XW,
                            const float* __restrict__ W_hh,
                            float* __restrict__ h_seq,
                            float* __restrict__ c_seq, int t) {
  const int lane = threadIdx.x & 31, wid = threadIdx.x >> 5;
  const int ll = lane & 15, lh = lane >> 4;
  const int tile = blockIdx.x * 8 + wid;      // 0..63
  const int m0 = (tile & 1) * 16;             // batch tile
  const int n0 = (tile >> 1) * 16;            // h-column tile

  const float* hprev = h_seq + (size_t)t * (Bn * Hn);
  const float* aptr  = hprev + (size_t)(m0 + ll) * Hn;
  const float* wi = W_hh + (size_t)(n0 + ll) * Hn;
  const float* wf = wi + (size_t)Hn * Hn;       // +512 rows
  const float* wg = wi + (size_t)2 * Hn * Hn;
  const float* wo = wi + (size_t)3 * Hn * Hn;

  v8f ai = {}, af = {}, ag = {}, ao = {};
  for (int k = 0; k < Hn; k += 4) {
    const int kk = k + 2 * lh;
    v2f a  = { aptr[kk], aptr[kk + 1] };
    v2f fi = { wi[kk], wi[kk + 1] };
    v2f ff = { wf[kk], wf[kk + 1] };
    v2f fg = { wg[kk], wg[kk + 1] };
    v2f fo = { wo[kk], wo[kk + 1] };
    ai = wmma4(a, fi, ai);
    af = wmma4(a, ff, af);
    ag = wmma4(a, fg, ag);
    ao = wmma4(a, fo, ao);
  }

  const float* cprev = c_seq + (size_t)t * (Bn * Hn);
  float* hnext = h_seq + (size_t)(t + 1) * (Bn * Hn);
  float* cnext = c_seq + (size_t)(t + 1) * (Bn * Hn);
  const int col = n0 + ll;
#pragma unroll
  for (int r = 0; r < 8; ++r) {
    const int b = m0 + r + 8 * lh;
    const size_t xr = ((size_t)b * Tn + t) * H4;
    const float gi = ai[r] + XW[xr + col];
    const float gf = af[r] + XW[xr + Hn + col];
    const float gg = ag[r] + XW[xr + 2 * Hn + col];
    const float go = ao[r] + XW[xr + 3 * Hn + col];
    const float cn = sigm(gf) * cprev[(size_t)b * Hn + col] + sigm(gi) * tanhf(gg);
    const float hn = sigm(go) * tanhf(cn);
    cnext[(size_t)b * Hn + col] = cn;
    hnext[(size_t)b * Hn + col] = hn;
  }
}

// ---------------------------------------------------------------------------
// Kernel 3 (per step t): attention + combine. One block per 16 batch rows.
// Phase1: q = h @ W_att^T (WMMA) -> LDS.  Phase2: scores w = q . enc (VALU).
// Phase3: applied = w . enc -> LDS (overwrites q).  Phase4: out =
// tanh([h|applied] @ W_comb^T + b) (WMMA, K=1536).
// grid 2 x 512 threads, dynamic LDS = (16*1024 + 16*64)*4 = 69632 B.
// ---------------------------------------------------------------------------
__global__ __launch_bounds__(512)
void k_attn_comb(const float* __restrict__ h_seq, const float* __restrict__ enc,
                 const float* __restrict__ W_att, const float* __restrict__ W_comb,
                 const float* __restrict__ b_comb, float* __restrict__ outs, int t) {
  extern __shared__ float smem[];
  float* qs = smem;                 // [16][1024] (q, later applied)
  float* ws = smem + 16 * H2;       // [16][64]
  const int tid = threadIdx.x;
  const int lane = tid & 31, wid = tid >> 5;
  const int ll = lane & 15, lh = lane >> 4;
  const int m0 = blockIdx.x * 16;
  const float* h = h_seq + (size_t)(t + 1) * (Bn * Hn);

  // -------- Phase 1: q tile [16 x 1024], wave wid owns cols wid*64..+63
  {
    const float* ap = h + (size_t)(m0 + ll) * Hn;
    const int nb = wid * 64;
    const float* b0 = W_att + (size_t)(nb +  0 + ll) * Hn;
    const float* b1 = W_att + (size_t)(nb + 16 + ll) * Hn;
    const float* b2 = W_att + (size_t)(nb + 32 + ll) * Hn;
    const float* b3 = W_att + (size_t)(nb + 48 + ll) * Hn;
    v8f a0 = {}, a1 = {}, a2 = {}, a3 = {};
    for (int k = 0; k < Hn; k += 4) {
      const int kk = k + 2 * lh;
      v2f a  = { ap[kk], ap[kk + 1] };
      v2f f0 = { b0[kk], b0[kk + 1] };
      v2f f1 = { b1[kk], b1[kk + 1] };
      v2f f2 = { b2[kk], b2[kk + 1] };
      v2f f3 = { b3[kk], b3[kk + 1] };
      a0 = wmma4(a, f0, a0);
      a1 = wmma4(a, f1, a1);
      a2 = wmma4(a, f2, a2);
      a3 = wmma4(a, f3, a3);
    }
#pragma unroll
    for (int i = 0; i < 4; ++i) {
      const v8f acc = (i == 0) ? a0 : (i == 1) ? a1 : (i == 2) ? a2 : a3;
#pragma unroll
      for (int r = 0; r < 8; ++r)
        qs[(r + 8 * lh) * H2 + nb + i * 16 + ll] = acc[r];
    }
  }
  __syncthreads();

  // -------- Phase 2: w[row][s] = q[row] . enc[b][s]  (two per thread)
  for (int idx = tid; idx < 16 * Sn; idx += 512) {
    const int row = idx >> 6, s = idx & 63;
    const float4* q4 = (const float4*)(qs + row * H2);
    const float4* e4 = (const float4*)(enc + ((size_t)(m0 + row) * Sn + s) * H2);
    float acc = 0.f;
    for (int j = 0; j < H2 / 4; ++j) {
      const float4 qa = q4[j], eb = e4[j];
      acc += qa.x * eb.x + qa.y * eb.y + qa.z * eb.z + qa.w * eb.w;
    }
    ws[idx] = acc;
  }
  __syncthreads();

  // -------- Phase 3: applied[row][j] = sum_s w[row][s]*enc[b][s][j] -> qs
  {
    const int row = tid >> 5;           // 0..15
    const int j0 = (tid & 31) * 32;     // 32 contiguous j per thread
    float acc[32];
#pragma unroll
    for (int j = 0; j < 32; ++j) acc[j] = 0.f;
    const float* e = enc + ((size_t)(m0 + row) * Sn) * H2 + j0;
    for (int s = 0; s < Sn; ++s) {
      const float wv = ws[row * Sn + s];
      const float* ep = e + (size_t)s * H2;
#pragma unroll
      for (int j = 0; j < 32; ++j) acc[j] += wv * ep[j];
    }
#pragma unroll
    for (int j = 0; j < 32; ++j) qs[row * H2 + j0 + j] = acc[j];
  }
  __syncthreads();

  // -------- Phase 4: out = tanh([h | applied] @ W_comb^T + b_comb)
  {
    const float* hrow = h + (size_t)(m0 + ll) * Hn;   // k < 512
    const float* arow = qs + ll * H2;                 // k >= 512 (applied)
#pragma unroll
    for (int tt = 0; tt < 2; ++tt) {
      const int n0 = (wid * 2 + tt) * 16;             // 32 tiles over H=512
      const float* bp = W_comb + (size_t)(n0 + ll) * H3;
      v8f acc = {};
      for (int k = 0; k < H3; k += 4) {
        const int kk = k + 2 * lh;                    // pair never straddles 512
        v2f a;
        a.x = (kk < Hn) ? hrow[kk] : arow[kk - Hn];
        a.y = (kk + 1 < Hn) ? hrow[kk + 1] : arow[kk + 1 - Hn];
        v2f f = { bp[kk], bp[kk + 1] };
        acc = wmma4(a, f, acc);
      }
      const int col = n0 + ll;
      const float bias = b_comb[col];
#pragma unroll
      for (int r = 0; r < 8; ++r) {
        const int b = m0 + r + 8 * lh;
        outs[((size_t)b * Tn + t) * Hn + col] = tanhf(acc[r] + bias);
      }
    }
  }
}

// ---------------------------------------------------------------------------
// Kernel 4: logits = outs @ W_out^T + b_out.  M=2048, N=32000, K=512.
// Wave computes a 32x64 tile: 6 b64 loads -> 8 WMMAs per k-step
// (A frags reused 4x, B frags 2x; ~10.7 matrix-FLOP per cache byte).
// grid (ceil(N/512)=63, M/32=64), block 256.
// ---------------------------------------------------------------------------
__global__ void k_logits(const float* __restrict__ outs,
                         const float* __restrict__ W_out,
                         const float* __restrict__ b_out,
                         float* __restrict__ logits) {
  const int lane = threadIdx.x & 31, wid = threadIdx.x >> 5;
  const int ll = lane & 15, lh = lane >> 4;
  const int m0 = blockIdx.y * 32;
  const int n0 = blockIdx.x * 512 + wid * 64;
  if (n0 >= Vn) return;                       // tail block: waves 4..7 idle

  const float* a0p = outs + (size_t)(m0 + ll) * Hn;
  const float* a1p = outs + (size_t)(m0 + 16 + ll) * Hn;
  const float* b0 = W_out + (size_t)(n0 +  0 + ll) * Hn;
  const float* b1 = W_out + (size_t)(n0 + 16 + ll) * Hn;
  const float* b2 = W_out + (size_t)(n0 + 32 + ll) * Hn;
  const float* b3 = W_out + (size_t)(n0 + 48 + ll) * Hn;

  v8f c00 = {}, c01 = {}, c02 = {}, c03 = {};
  v8f c10 = {}, c11 = {}, c12 = {}, c13 = {};
  for (int k = 0; k < Hn; k += 4) {
    const int kk = k + 2 * lh;
    v2f a0 = { a0p[kk], a0p[kk + 1] };
    v2f a1 = { a1p[kk], a1p[kk + 1] };
    v2f f0 = { b0[kk], b0[kk + 1] };
    v2f f1 = { b1[kk], b1[kk + 1] };
    v2f f2 = { b2[kk], b2[kk + 1] };
    v2f f3 = { b3[kk], b3[kk + 1] };
    c00 = wmma4(a0, f0, c00);  c01 = wmma4(a0, f1, c01);
    c02 = wmma4(a0, f2, c02);  c03 = wmma4(a0, f3, c03);
    c10 = wmma4(a1, f0, c10);  c11 = wmma4(a1, f1, c11);
    c12 = wmma4(a1, f2, c12);  c13 = wmma4(a1, f3, c13);
  }
#pragma unroll
  for (int i = 0; i < 4; ++i) {
    const v8f lo = (i == 0) ? c00 : (i == 1) ? c01 : (i == 2) ? c02 : c03;
    const v8f hi = (i == 0) ? c10 : (i == 1) ? c11 : (i == 2) ? c12 : c13;
    const int col = n0 + i * 16 + ll;
    const float bias = b_out[col];
#pragma unroll
    for (int r = 0; r < 8; ++r) {
      logits[(size_t)(m0 + r + 8 * lh) * Vn + col] = lo[r] + bias;
      logits[(size_t)(m0 + 16 + r + 8 * lh) * Vn + col] = hi[r] + bias;
    }
  }
}

// ---------------------------------------------------------------------------
// Kernel 5: append final h, c to d_out tail (return order: logits, h, c)
// ---------------------------------------------------------------------------
__global__ void k_final(const float* __restrict__ h_seq,
                        const float* __restrict__ c_seq,
                        float* __restrict__ out_tail) {
  int i = blockIdx.x * blockDim.x + threadIdx.x;
  if (i < Bn * Hn) {
    out_tail[i] = h_seq[(size_t)Tn * Bn * Hn + i];
    out_tail[Bn * Hn + i] = c_seq[(size_t)Tn * Bn * Hn + i];
  }
}

// ---------------------------------------------------------------------------
extern "C" void kernel_launch(void* const* d_in, const int* in_sizes, int n_in,
                              void* d_out, int out_size, void* d_ws, size_t ws_size,
                              hipStream_t stream) {
  const int*   inputs = (const int*)  d_in[0];
  const float* enc    = (const float*)d_in[1];
  const float* h0     = (const float*)d_in[2];
  const float* c0     = (const float*)d_in[3];
  const float* emb    = (const float*)d_in[4];
  const float* W_ih   = (const float*)d_in[5];
  const float* W_hh   = (const float*)d_in[6];
  const float* b_ih   = (const float*)d_in[7];
  const float* b_hh   = (const float*)d_in[8];
  const float* W_att  = (const float*)d_in[9];
  const float* W_comb = (const float*)d_in[10];
  const float* b_comb = (const float*)d_in[11];
  const float* W_out  = (const float*)d_in[12];
  const float* b_out  = (const float*)d_in[13];

  float* ws    = (float*)d_ws;
  float* XW    = ws;                               // 2048*2048
  float* h_seq = XW + (size_t)MT * H4;             // 65*32*512
  float* c_seq = h_seq + (size_t)(Tn + 1) * Bn * Hn;
  float* outs  = c_seq + (size_t)(Tn + 1) * Bn * Hn;  // 2048*512
  float* logits = (float*)d_out;

  k_init<<<64, 256, 0, stream>>>(h0, c0, h_seq, c_seq);
  k_embed_xw<<<dim3(4, 64), 256, 0, stream>>>(inputs, emb, W_ih, b_ih, b_hh, XW);
  for (int t = 0; t < Tn; ++t) {
    k_lstm_step<<<8, 256, 0, stream>>>(XW, W_hh, h_seq, c_seq, t);
    k_attn_comb<<<2, 512, (16 * H2 + 16 * Sn) * sizeof(float), stream>>>(
        h_seq, enc, W_att, W_comb, b_comb, outs, t);
  }
  k_logits<<<dim3(63, 64), 256, 0, stream>>>(outs, W_out, b_out, logits);
  k_final<<<64, 256, 0, stream>>>(h_seq, c_seq, logits + (size_t)MT * Vn);
}